// Network_61615600828839
// MI455X (gfx1250) — compile-verified
//
#include <hip/hip_runtime.h>

// FWHT-4096 of (in1 * in2) per row, scaled by 1/64, one row per 256-thread block.
// H4096 = H16 (x) H16 (x) H16 ; each H16 contraction done with
// V_WMMA_F32_16X16X4_F32 (fp32 WMMA, 4 chained K=4 steps per 16x16 tile).

typedef __attribute__((ext_vector_type(2))) float v2f;
typedef __attribute__((ext_vector_type(8))) float v8f;

#define FWHT_N 4096
// 4 floats of padding per 16-element group: keeps float4 LDS ops 16B-aligned
// and spreads the strided pass accesses across all 64 banks (stride 20).
#define LDS_FLOATS (4096 + 1024)

__device__ __forceinline__ int pidx(int e) { return e + ((e >> 4) << 2); }

// Natural-order Hadamard coefficient: H16[a,b] = (-1)^popcount(a&b)
__device__ __forceinline__ float hsign(int a, int b) {
    return (__builtin_popcount(a & b) & 1) ? -1.0f : 1.0f;
}

// One 16x16 tile of one contraction pass, done by one wave32.
//   D[m, n] = sum_K H16[m, K] * X[f*Sf + K*Sc + n*Sn]   (written back in place)
// A-matrix (16x4 f32): lane L holds A[L%16, 2*(L/16)+v] in vgpr v.
// B-matrix (4x16 f32): lane L holds B[2*(L/16)+v, L%16] in vgpr v.
// D (16x16 f32):       lane L holds D[v+8*(L/16), L%16] in vgpr v.
__device__ __forceinline__ void hadamard16_tile(float* lds, int lane, int f,
                                                int Sf, int Sc, int Sn) {
    const int n    = lane & 15;
    const int hi   = lane >> 4;
    const int base = f * Sf + n * Sn;
    v8f acc = {};
#pragma unroll
    for (int k0 = 0; k0 < 16; k0 += 4) {
        v2f A, Bv;
#pragma unroll
        for (int v = 0; v < 2; ++v) {
            const int K = k0 + hi * 2 + v;
            A[v]  = hsign(n, K);
            Bv[v] = lds[pidx(base + K * Sc)];
        }
        // (neg_a, A, neg_b, B, c_mod, C, reuse_a, reuse_b)
        acc = __builtin_amdgcn_wmma_f32_16x16x4_f32(false, A, false, Bv,
                                                    (short)0, acc, false, false);
    }
#pragma unroll
    for (int v = 0; v < 8; ++v) {
        const int m = v + 8 * hi;
        lds[pidx(f * Sf + m * Sc + n * Sn)] = acc[v];
    }
}

__global__ __launch_bounds__(256)
void fwht_mul_kernel(const float* __restrict__ in1,
                     const float* __restrict__ in2,
                     float* __restrict__ out) {
    __shared__ float lds[LDS_FLOATS];
    const int t    = threadIdx.x;
    const int lane = t & 31;
    const int wv   = t >> 5;                       // 8 waves per block
    const size_t rowoff = (size_t)blockIdx.x * FWHT_N;
    const float* p1 = in1 + rowoff;
    const float* p2 = in2 + rowoff;
    float*       po = out + rowoff;

    // Fused elementwise multiply while staging the row into (padded) LDS.
#pragma unroll
    for (int q = 0; q < 4; ++q) {
        const int e = (q * 256 + t) * 4;           // 16B-aligned within a group
        const float4 a = *(const float4*)(p1 + e);
        const float4 b = *(const float4*)(p2 + e);
        float4 c;
        c.x = a.x * b.x; c.y = a.y * b.y; c.z = a.z * b.z; c.w = a.w * b.w;
        *(float4*)(&lds[pidx(e)]) = c;
    }
    __syncthreads();

    // Pass 1: contract k (stride 1); tiles over i (stride 256); free axis j (16).
    hadamard16_tile(lds, lane, wv,     256, 1, 16);
    hadamard16_tile(lds, lane, wv + 8, 256, 1, 16);
    __syncthreads();
    // Pass 2: contract j (stride 16); tiles over i (256); free axis k (1).
    hadamard16_tile(lds, lane, wv,     256, 16, 1);
    hadamard16_tile(lds, lane, wv + 8, 256, 16, 1);
    __syncthreads();
    // Pass 3: contract i (stride 256); tiles over j (16); free axis k (1).
    hadamard16_tile(lds, lane, wv,     16, 256, 1);
    hadamard16_tile(lds, lane, wv + 8, 16, 256, 1);
    __syncthreads();

    // Scale by 1/sqrt(4096) on the way out.
    const float scale = 1.0f / 64.0f;
#pragma unroll
    for (int q = 0; q < 4; ++q) {
        const int e = (q * 256 + t) * 4;
        float4 r = *(const float4*)(&lds[pidx(e)]);
        r.x *= scale; r.y *= scale; r.z *= scale; r.w *= scale;
        *(float4*)(po + e) = r;
    }
}

extern "C" void kernel_launch(void* const* d_in, const int* in_sizes, int n_in,
                              void* d_out, int out_size, void* d_ws, size_t ws_size,
                              hipStream_t stream) {
    const float* in1 = (const float*)d_in[0];
    const float* in2 = (const float*)d_in[1];
    float* out = (float*)d_out;
    const int rows = in_sizes[0] / FWHT_N;         // 8192 for B=8192, N=4096
    fwht_mul_kernel<<<rows, 256, 0, stream>>>(in1, in2, out);
}